// TTTLinearAdaptation_84524956385847
// MI455X (gfx1250) — compile-verified
//
#include <hip/hip_runtime.h>
#include <math.h>

#if defined(__has_builtin)
#if __has_builtin(__builtin_amdgcn_global_load_async_to_lds_b128)
#define HAVE_ASYNC_LDS 1
#endif
#if __has_builtin(__builtin_amdgcn_s_wait_asynccnt)
#define HAVE_WAIT_ASYNC 1
#endif
#endif
#ifndef HAVE_ASYNC_LDS
#define HAVE_ASYNC_LDS 0
#endif
#ifndef HAVE_WAIT_ASYNC
#define HAVE_WAIT_ASYNC 0
#endif

namespace {
constexpr int NH   = 16;    // heads
constexpr int HD   = 64;    // head dim
constexpr int CH   = 16;    // chunk
constexpr int BB   = 4;     // batch
constexpr int SEQ  = 8192;
constexpr int NCH  = SEQ / CH;
constexpr float LN_EPS = 1e-6f;

constexpr int WSTR = 68;    // LDS row stride for W (padding vs bank conflicts)
constexpr int TSTR = 68;    // LDS row stride for token-major 16x64 tiles

typedef __attribute__((ext_vector_type(2))) float v2f;
typedef __attribute__((ext_vector_type(8))) float v8f;

__device__ __forceinline__ v8f wmma4(v2f a, v2f b, v8f c) {
  // V_WMMA_F32_16X16X4_F32 : D(16x16 f32) = A(16x4 f32) * B(4x16 f32) + C
  return __builtin_amdgcn_wmma_f32_16x16x4_f32(false, a, false, b, (short)0, c,
                                               false, false);
}

// One 16-byte global -> LDS transfer. Async (ASYNCcnt-tracked) when the
// gfx1250 builtin is available; plain copy otherwise (still double-buffered).
__device__ __forceinline__ void async_copy_b128(const float* gsrc, float* ldst) {
#if HAVE_ASYNC_LDS
  typedef int v4i __attribute__((vector_size(16)));
  typedef __attribute__((address_space(1))) v4i* gvp_t;   // "__device__" (global)
  typedef __attribute__((address_space(3))) v4i* lvp_t;   // "__shared__" (LDS)
  __builtin_amdgcn_global_load_async_to_lds_b128((gvp_t)gsrc, (lvp_t)ldst, 0, 0);
#else
  *(float4*)ldst = *(const float4*)gsrc;
#endif
}

__device__ __forceinline__ void wait_async() {
#if HAVE_ASYNC_LDS
#if HAVE_WAIT_ASYNC
  __builtin_amdgcn_s_wait_asynccnt(0);
#else
  asm volatile("s_wait_asynccnt 0" ::: "memory");
#endif
#endif
}
} // namespace

__launch_bounds__(128, 1)
__global__ void ttt_linear_scan(const float* __restrict__ xq,
                                const float* __restrict__ xk,
                                const float* __restrict__ xv,
                                const float* __restrict__ W0,
                                const float* __restrict__ b0,
                                const float* __restrict__ gamma,
                                const float* __restrict__ beta,
                                const float* __restrict__ theta,
                                const float* __restrict__ theta_bias,
                                const float* __restrict__ alpha,
                                float* __restrict__ out)
{
  __shared__ float W_s [HD * WSTR];        // fast-weight state, row-major [d][e]
  __shared__ float kbuf[2][CH * TSTR];     // double-buffered token-major [c][d]
  __shared__ float qbuf[2][CH * TSTR];
  __shared__ float vbuf[2][CH * TSTR];
  __shared__ float zbuf[CH * TSTR];        // Z, later Zq
  __shared__ float gxbuf[CH * TSTR];       // grad_x
  __shared__ float b_s[HD], gam_s[HD], bet_s[HD], th_s[HD];
  __shared__ float mu_s[CH], rstd_s[CH], sg_s[CH], sgx_s[CH], eta_s[CH], tok_s[CH];
  __shared__ float thb_s;

  const int tid  = threadIdx.x;
  const int lane = tid & 31;
  const int wv   = tid >> 5;     // wave id = owned column block j (0..3)
  const int m    = lane & 15;    // M for A operands / N for B & C operands
  const int hl   = lane >> 4;    // lane half (0/1)

  const int bh = blockIdx.x;     // b*NH + h
  const int h  = bh % NH;

  // ---------------- init per-sequence state ----------------
  {
    const float* w0 = W0 + (size_t)h * HD * HD;
    #pragma unroll
    for (int i = 0; i < 8; ++i) {
      int off = (tid + 128 * i) * 4;                // 0..4095
      float4 w = *(const float4*)(w0 + off);
      int row = off >> 6, col = off & 63;
      *(float4*)&W_s[row * WSTR + col] = w;
    }
    if (tid < HD) {
      b_s[tid]   = b0[h * HD + tid];
      gam_s[tid] = gamma[h * HD + tid];
      bet_s[tid] = beta[h * HD + tid];
      th_s[tid]  = theta[h * HD + tid];
    }
    if (tid < CH) tok_s[tid] = fmaxf(1.0f / (float)(tid + 1) + alpha[tid], 0.0f);
    if (tid == 0) thb_s = theta_bias[h];
  }

  const size_t seqbase = (size_t)bh * SEQ * HD;

  // stage chunk 0 into buffer 0 (async engine runs while we hit the barrier)
  {
    #pragma unroll
    for (int i = 0; i < 2; ++i) {
      int off = (tid + 128 * i) * 4;                // 0..1023
      int row = off >> 6, col = off & 63;
      async_copy_b128(xq + seqbase + off, &qbuf[0][row * TSTR + col]);
      async_copy_b128(xk + seqbase + off, &kbuf[0][row * TSTR + col]);
      async_copy_b128(xv + seqbase + off, &vbuf[0][row * TSTR + col]);
    }
  }

  for (int cn = 0; cn < NCH; ++cn) {
    const int cur = cn & 1;
    wait_async();                                    // my async transfers done
    __syncthreads();                                 // (A) everyone's done; prev bufs free

    const size_t cbase = seqbase + (size_t)cn * CH * HD;
    const float* kc = kbuf[cur];
    const float* qc = qbuf[cur];
    const float* vc = vbuf[cur];

    // kick off next chunk into the other buffer; overlaps whole chunk compute
    if (cn + 1 < NCH) {
      const size_t nbase = cbase + CH * HD;
      float* qn = qbuf[cur ^ 1];
      float* kn = kbuf[cur ^ 1];
      float* vn = vbuf[cur ^ 1];
      #pragma unroll
      for (int i = 0; i < 2; ++i) {
        int off = (tid + 128 * i) * 4;
        int row = off >> 6, col = off & 63;
        async_copy_b128(xq + nbase + off, &qn[row * TSTR + col]);
        async_copy_b128(xk + nbase + off, &kn[row * TSTR + col]);
        async_copy_b128(xv + nbase + off, &vn[row * TSTR + col]);
      }
      __builtin_prefetch(xq + nbase + CH * HD + tid * 8, 0, 0);  // warm L2 for cn+2
      __builtin_prefetch(xk + nbase + CH * HD + tid * 8, 0, 0);
      __builtin_prefetch(xv + nbase + CH * HD + tid * 8, 0, 0);
    }

    // ---------------- Z = k @ W + b (wave wv -> columns 16wv..16wv+15) ------
    v8f acc;
    {
      float bval = b_s[16 * wv + m];
      #pragma unroll
      for (int r = 0; r < 8; ++r) acc[r] = bval;
      #pragma unroll
      for (int kb = 0; kb < 16; ++kb) {              // K = 64 in steps of 4
        int d0 = 4 * kb + 2 * hl;
        v2f a, bm;
        a.x  = kc[m * TSTR + d0];
        a.y  = kc[m * TSTR + d0 + 1];
        bm.x = W_s[d0 * WSTR + 16 * wv + m];
        bm.y = W_s[(d0 + 1) * WSTR + 16 * wv + m];
        acc  = wmma4(a, bm, acc);
      }
      #pragma unroll
      for (int r = 0; r < 8; ++r)
        zbuf[(r + 8 * hl) * TSTR + 16 * wv + m] = acc[r];
    }
    __syncthreads();                                 // (B) Z ready

    // ------- per-token LN stats + eta: 8 lanes per token, shfl_xor reduce ---
    {
      const int t   = tid >> 3;                      // token 0..15
      const int sub = tid & 7;                       // 8 features each
      const float* zr = &zbuf[t * TSTR];
      const float* kr = &kc[t * TSTR];
      const float* vr = &vc[t * TSTR];
      float s = 0.f, ss = 0.f, dot = 0.f;
      #pragma unroll
      for (int i = 0; i < 8; ++i) {
        int e = sub * 8 + i;
        float z = zr[e];
        s += z; ss += z * z;
        dot += kr[e] * th_s[e];
      }
      #pragma unroll
      for (int msk = 1; msk < 8; msk <<= 1) {
        s   += __shfl_xor(s,   msk, 32);
        ss  += __shfl_xor(ss,  msk, 32);
        dot += __shfl_xor(dot, msk, 32);
      }
      float mu   = s * (1.0f / HD);
      float var  = ss * (1.0f / HD) - mu * mu;
      float rstd = 1.0f / sqrtf(var + LN_EPS);
      float sg = 0.f, sgx = 0.f;
      #pragma unroll
      for (int i = 0; i < 8; ++i) {
        int e = sub * 8 + i;
        float xh = (zr[e] - mu) * rstd;
        float g  = (gam_s[e] * xh + bet_s[e] - vr[e] + kr[e]) * gam_s[e];
        sg += g; sgx += g * xh;
      }
      #pragma unroll
      for (int msk = 1; msk < 8; msk <<= 1) {
        sg  += __shfl_xor(sg,  msk, 32);
        sgx += __shfl_xor(sgx, msk, 32);
      }
      if (sub == 0) {
        float lr = 1.0f / (1.0f + expf(-(dot + thb_s)));
        mu_s[t] = mu; rstd_s[t] = rstd; sg_s[t] = sg; sgx_s[t] = sgx;
        eta_s[t] = (1.0f / HD) * lr * tok_s[t];
      }
    }
    __syncthreads();                                 // (C)

    // ---------------- grad_x (elementwise, all threads) ---------------------
    #pragma unroll
    for (int i = 0; i < 8; ++i) {
      int idx = tid + 128 * i;                       // 0..1023
      int t = idx >> 6, e = idx & 63;
      float z  = zbuf[t * TSTR + e];
      float xh = (z - mu_s[t]) * rstd_s[t];
      float g  = (gam_s[e] * xh + bet_s[e] - vc[t * TSTR + e]
                  + kc[t * TSTR + e]) * gam_s[e];
      gxbuf[t * TSTR + e] =
          ((float)HD * g - sg_s[t] - xh * sgx_s[t]) * (rstd_s[t] * (1.0f / HD));
    }
    __syncthreads();                                 // (D) grad_x ready

    // -------- W -= (eta*k)^T @ grad_x : accumulate onto W as C operand ------
    #pragma unroll
    for (int i = 0; i < 4; ++i) {                    // row blocks of W
      v8f wacc;
      #pragma unroll
      for (int r = 0; r < 8; ++r)
        wacc[r] = W_s[(16 * i + r + 8 * hl) * WSTR + 16 * wv + m];
      #pragma unroll
      for (int kb = 0; kb < 4; ++kb) {               // K = 16 tokens, steps of 4
        int c0 = 4 * kb + 2 * hl;
        v2f a, bm;
        a.x  = -eta_s[c0]     * kc[c0 * TSTR + 16 * i + m];
        a.y  = -eta_s[c0 + 1] * kc[(c0 + 1) * TSTR + 16 * i + m];
        bm.x = gxbuf[c0 * TSTR + 16 * wv + m];
        bm.y = gxbuf[(c0 + 1) * TSTR + 16 * wv + m];
        wacc = wmma4(a, bm, wacc);
      }
      #pragma unroll
      for (int r = 0; r < 8; ++r)
        W_s[(16 * i + r + 8 * hl) * WSTR + 16 * wv + m] = wacc[r];
    }
    if (lane < 16) {                                 // b -= db on owned columns
      int e = 16 * wv + lane;
      float db = 0.f;
      #pragma unroll
      for (int c = 0; c < CH; ++c) db += eta_s[c] * gxbuf[c * TSTR + e];
      b_s[e] -= db;
    }

    // ---- Zq = q @ W_new + b_new (same-wave column ownership: no barrier) ---
    {
      float bval = b_s[16 * wv + m];
      #pragma unroll
      for (int r = 0; r < 8; ++r) acc[r] = bval;
      #pragma unroll
      for (int kb = 0; kb < 16; ++kb) {
        int d0 = 4 * kb + 2 * hl;
        v2f a, bm;
        a.x  = qc[m * TSTR + d0];
        a.y  = qc[m * TSTR + d0 + 1];
        bm.x = W_s[d0 * WSTR + 16 * wv + m];
        bm.y = W_s[(d0 + 1) * WSTR + 16 * wv + m];
        acc  = wmma4(a, bm, acc);
      }
      #pragma unroll
      for (int r = 0; r < 8; ++r)
        zbuf[(r + 8 * hl) * TSTR + 16 * wv + m] = acc[r];
    }
    __syncthreads();                                 // (E) Zq ready

    // ------- LN stats for Zq (same parallel reduction) ----------------------
    {
      const int t   = tid >> 3;
      const int sub = tid & 7;
      const float* zr = &zbuf[t * TSTR];
      float s = 0.f, ss = 0.f;
      #pragma unroll
      for (int i = 0; i < 8; ++i) {
        float z = zr[sub * 8 + i];
        s += z; ss += z * z;
      }
      #pragma unroll
      for (int msk = 1; msk < 8; msk <<= 1) {
        s  += __shfl_xor(s,  msk, 32);
        ss += __shfl_xor(ss, msk, 32);
      }
      if (sub == 0) {
        float mu  = s * (1.0f / HD);
        float var = ss * (1.0f / HD) - mu * mu;
        mu_s[t]   = mu;
        rstd_s[t] = 1.0f / sqrtf(var + LN_EPS);
      }
    }
    __syncthreads();                                 // (F)

    // ---------------- out = q + LN(Zq), coalesced b128 store ---------------
    #pragma unroll
    for (int i = 0; i < 2; ++i) {
      int off = (tid + 128 * i) * 4;
      int t = off >> 6, e = off & 63;
      float4 zq = *(const float4*)&zbuf[t * TSTR + e];
      float4 qq = *(const float4*)&qc[t * TSTR + e];
      float mu = mu_s[t], rstd = rstd_s[t];
      float4 o;
      o.x = qq.x + gam_s[e + 0] * ((zq.x - mu) * rstd) + bet_s[e + 0];
      o.y = qq.y + gam_s[e + 1] * ((zq.y - mu) * rstd) + bet_s[e + 1];
      o.z = qq.z + gam_s[e + 2] * ((zq.z - mu) * rstd) + bet_s[e + 2];
      o.w = qq.w + gam_s[e + 3] * ((zq.w - mu) * rstd) + bet_s[e + 3];
      *(float4*)(out + cbase + off) = o;
    }
  }
}

extern "C" void kernel_launch(void* const* d_in, const int* in_sizes, int n_in,
                              void* d_out, int out_size, void* d_ws, size_t ws_size,
                              hipStream_t stream) {
  (void)in_sizes; (void)n_in; (void)out_size; (void)d_ws; (void)ws_size;
  const float* xq    = (const float*)d_in[0];
  const float* xk    = (const float*)d_in[1];
  const float* xv    = (const float*)d_in[2];
  const float* W0    = (const float*)d_in[3];
  const float* b0    = (const float*)d_in[4];
  const float* gamma = (const float*)d_in[5];
  const float* beta  = (const float*)d_in[6];
  const float* theta = (const float*)d_in[7];
  const float* thb   = (const float*)d_in[8];
  const float* alpha = (const float*)d_in[9];
  float* out = (float*)d_out;

  dim3 grid(BB * NH);   // one workgroup per (batch, head) sequence
  dim3 block(128);      // 4 wave32s
  ttt_linear_scan<<<grid, block, 0, stream>>>(xq, xk, xv, W0, b0, gamma, beta,
                                              theta, thb, alpha, out);
}